// DGCNN_voxel_reshape_6227702579203
// MI455X (gfx1250) — compile-verified
//
#include <hip/hip_runtime.h>
#include <hip/hip_bf16.h>

// ---------------------------------------------------------------------------
// DGCNN forward for gfx1250 (CDNA5, wave32, WMMA f32 16x16x4).
// All GEMMs (edge convs, conv5/conv8, FC layers) run on V_WMMA_F32_16X16X4_F32
// keeping exact f32 math. conv5/conv8 are fused with global max+mean pooling
// so the (B,1024,512) activation is never materialized.
// ---------------------------------------------------------------------------

typedef float v2f __attribute__((ext_vector_type(2)));
typedef float v8f __attribute__((ext_vector_type(8)));

__device__ __forceinline__ v8f v8zero() {
  v8f z;
#pragma unroll
  for (int r = 0; r < 8; ++r) z[r] = 0.f;
  return z;
}

__device__ __forceinline__ v8f wmma4(v2f a, v2f b, v8f c) {
  // D(16x16) += A(16x4) * B(4x16), all f32.
  return __builtin_amdgcn_wmma_f32_16x16x4_f32(false, a, false, b, (short)0, c,
                                               false, false);
}

__device__ __forceinline__ float lrelu(float v) {
  return v >= 0.f ? v : 0.2f * v;
}

// ------------------------------- prep kernels ------------------------------

__global__ void bnprep_kernel(const float* __restrict__ g, const float* __restrict__ b,
                              const float* __restrict__ m, const float* __restrict__ v,
                              int C, float* __restrict__ s, float* __restrict__ h) {
  int i = blockIdx.x * blockDim.x + threadIdx.x;
  if (i < C) {
    float sc = g[i] * rsqrtf(v[i] + 1e-5f);
    s[i] = sc;
    h[i] = b[i] - m[i] * sc;
  }
}

// input (B,512,9) -> x0 (B,9,512)
__global__ void reshape_in_kernel(const float* __restrict__ in, float* __restrict__ x0,
                                  int total) {
  int i = blockIdx.x * blockDim.x + threadIdx.x;
  if (i >= total) return;
  int n = i % 512;
  int c = (i / 512) % 9;
  int b = i / (512 * 9);
  x0[i] = in[((size_t)(b * 512 + n)) * 9 + c];
}

// cls (64,40) -> y0 (2,40,32)
__global__ void reshape_cls_kernel(const float* __restrict__ cls, float* __restrict__ y0,
                                   int total) {
  int i = blockIdx.x * blockDim.x + threadIdx.x;
  if (i >= total) return;
  int v = i % 32;
  int c = (i / 32) % 40;
  int g = i / (32 * 40);
  y0[i] = cls[((size_t)(g * 32 + v)) * 40 + c];
}

// xx[b][n] = sum_c x[b][c][n]^2
__global__ void sqnorm_kernel(const float* __restrict__ x, int bstride, int C, int N,
                              float* __restrict__ xx, int total) {
  int i = blockIdx.x * blockDim.x + threadIdx.x;
  if (i >= total) return;
  int n = i % N;
  int b = i / N;
  const float* xb = x + (size_t)b * bstride;
  float s = 0.f;
  for (int c = 0; c < C; ++c) {
    float t = xb[c * N + n];
    s += t * t;
  }
  xx[i] = s;
}

// ----------------------------------- KNN -----------------------------------
// One thread per (b,n). pd[n][m] = -||x_n - x_m||^2 * (xx[m]>0 ? 1 : 1e7).
// Register top-K insertion chain; strict '>' keeps lower indices first on ties
// (matches jax.lax.top_k since m scans ascending).

template <int KK>
__global__ void knn_kernel(const float* __restrict__ x, int bstride, int C, int N,
                           const float* __restrict__ xx, int* __restrict__ idx,
                           int total) {
  int i = blockIdx.x * blockDim.x + threadIdx.x;
  if (i >= total) return;
  int n = i % N;
  int b = i / N;
  const float* xb = x + (size_t)b * bstride;
  const float* xxb = xx + (size_t)b * N;
  float bestv[KK];
  int besti[KK];
#pragma unroll
  for (int s = 0; s < KK; ++s) {
    bestv[s] = -3.4e38f;
    besti[s] = 0;
  }
  for (int m = 0; m < N; ++m) {
    float sq = 0.f;
    for (int c = 0; c < C; ++c) {
      float d = xb[c * N + n] - xb[c * N + m];
      sq += d * d;
    }
    float v = -sq * (xxb[m] > 0.f ? 1.f : 1e7f);
    int id = m;
#pragma unroll
    for (int s = 0; s < KK; ++s) {
      if (v > bestv[s]) {
        float tv = bestv[s];
        int ti = besti[s];
        bestv[s] = v;
        besti[s] = id;
        v = tv;
        id = ti;
      }
    }
  }
  int* ob = idx + (size_t)i * KK;
#pragma unroll
  for (int s = 0; s < KK; ++s) ob[s] = besti[s];
}

// -------------------------------- edge conv --------------------------------
// One wave per (b, point n). Columns of the WMMA tile = K neighbors
// (KNB=8 duplicates columns; max over duplicates is unchanged).
// out[b][o][n] = max_j lrelu( bn( sum_c W[o][c] * F[c][j] ) )
// F[c][j] = c<C ? x[b][c][idx_j] - x[b][c][n] : x[b][c-C][n]

template <int KNB>
__global__ void edgeconv_kernel(const float* __restrict__ xin, int in_bstride, int C,
                                int N, const int* __restrict__ idx,
                                const float* __restrict__ W, int O,
                                const float* __restrict__ bns,
                                const float* __restrict__ bnh, float* __restrict__ out,
                                int out_bstride) {
  int n = blockIdx.x;
  int b = blockIdx.y;
  int lane = threadIdx.x;
  int col = lane & 15;
  int klo = (lane < 16) ? 0 : 2;
  int twoC = 2 * C;
  const float* xb = xin + (size_t)b * in_bstride;
  int j = idx[((size_t)b * N + n) * KNB + (col & (KNB - 1))];
  int numMT = O >> 4;
  for (int mg = 0; mg < numMT; mg += 8) {
    int gcnt = numMT - mg;
    if (gcnt > 8) gcnt = 8;
    v8f acc[8];
#pragma unroll
    for (int g = 0; g < 8; ++g) acc[g] = v8zero();
    for (int kk = 0; kk < twoC; kk += 4) {
      v2f bf;
#pragma unroll
      for (int t = 0; t < 2; ++t) {
        int c = kk + klo + t;
        float bv = 0.f;
        if (c < C)
          bv = xb[c * N + j] - xb[c * N + n];
        else if (c < twoC)
          bv = xb[(c - C) * N + n];
        bf[t] = bv;
      }
#pragma unroll
      for (int g = 0; g < 8; ++g) {
        if (g < gcnt) {
          int o = (mg + g) * 16 + col;
          v2f af;
#pragma unroll
          for (int t = 0; t < 2; ++t) {
            int c = kk + klo + t;
            af[t] = (c < twoC) ? W[(size_t)o * twoC + c] : 0.f;
          }
          acc[g] = wmma4(af, bf, acc[g]);
        }
      }
    }
#pragma unroll
    for (int g = 0; g < 8; ++g) {
      if (g < gcnt) {
#pragma unroll
        for (int r = 0; r < 8; ++r) {
          int o = (mg + g) * 16 + r + ((lane < 16) ? 0 : 8);
          float v = acc[g][r];
          if (bns) v = v * bns[o] + bnh[o];
          v = lrelu(v);
          v = fmaxf(v, __shfl_xor(v, 8, 16));
          v = fmaxf(v, __shfl_xor(v, 4, 16));
          v = fmaxf(v, __shfl_xor(v, 2, 16));
          v = fmaxf(v, __shfl_xor(v, 1, 16));
          if (col == 0) out[(size_t)b * out_bstride + o * N + n] = v;
        }
      }
    }
  }
}

// ------------------------- conv + global pool (fused) ----------------------
// One wave per (b, 16-row output tile). Runs the 1x1 conv over all N columns
// in 16-wide WMMA tiles, applying bn+lrelu and accumulating running max/sum;
// never materializes the (B,O,N) activation.
// feat[b][0:O] = max_n, feat[b][O:2O] = mean_n.

__global__ void convpool_kernel(const float* __restrict__ xin, int C, int N,
                                const float* __restrict__ W,
                                const float* __restrict__ bns,
                                const float* __restrict__ bnh,
                                float* __restrict__ feat, int O) {
  int ot = blockIdx.x;
  int b = blockIdx.y;
  int lane = threadIdx.x;
  int col = lane & 15;
  int klo = (lane < 16) ? 0 : 2;
  const float* xb = xin + (size_t)b * C * N;
  int orow = ot * 16 + col;
  float vmax[8], vsum[8];
#pragma unroll
  for (int r = 0; r < 8; ++r) {
    vmax[r] = -3.4e38f;
    vsum[r] = 0.f;
  }
  for (int n0 = 0; n0 < N; n0 += 16) {
    v8f acc = v8zero();
    for (int kk = 0; kk < C; kk += 4) {
      v2f af, bf;
#pragma unroll
      for (int t = 0; t < 2; ++t) {
        int c = kk + klo + t;
        af[t] = W[(size_t)orow * C + c];
        bf[t] = xb[(size_t)c * N + n0 + col];
      }
      acc = wmma4(af, bf, acc);
    }
#pragma unroll
    for (int r = 0; r < 8; ++r) {
      int o = ot * 16 + r + ((lane < 16) ? 0 : 8);
      float v = acc[r] * bns[o] + bnh[o];
      v = lrelu(v);
      vmax[r] = fmaxf(vmax[r], v);
      vsum[r] += v;
    }
  }
#pragma unroll
  for (int r = 0; r < 8; ++r) {
    float m = vmax[r];
    float s = vsum[r];
    m = fmaxf(m, __shfl_xor(m, 8, 16));
    s += __shfl_xor(s, 8, 16);
    m = fmaxf(m, __shfl_xor(m, 4, 16));
    s += __shfl_xor(s, 4, 16);
    m = fmaxf(m, __shfl_xor(m, 2, 16));
    s += __shfl_xor(s, 2, 16);
    m = fmaxf(m, __shfl_xor(m, 1, 16));
    s += __shfl_xor(s, 1, 16);
    int o = ot * 16 + r + ((lane < 16) ? 0 : 8);
    if (col == 0) {
      feat[(size_t)b * 2 * O + o] = m;
      feat[(size_t)b * 2 * O + O + o] = s / (float)N;
    }
  }
}

// ------------------------------ linear (GEMM) ------------------------------
// Y[m][o] = epilogue( sum_c X[m][c] * W[o][c] ). One wave per 16x16 D tile.

__global__ void linear_kernel(const float* __restrict__ X, int Brows, int IN,
                              const float* __restrict__ W,
                              const float* __restrict__ bias,
                              const float* __restrict__ bns,
                              const float* __restrict__ bnh, int do_lrelu,
                              float* __restrict__ Y, int OUT) {
  int ot = blockIdx.x;
  int bt = blockIdx.y;
  int lane = threadIdx.x;
  int col = lane & 15;
  int klo = (lane < 16) ? 0 : 2;
  int row = bt * 16 + col;  // A-matrix row for loads
  int oc = ot * 16 + col;   // B-matrix column for loads
  v8f acc = v8zero();
  for (int kk = 0; kk < IN; kk += 4) {
    v2f af, bf;
#pragma unroll
    for (int t = 0; t < 2; ++t) {
      int c = kk + klo + t;
      af[t] = (row < Brows) ? X[(size_t)row * IN + c] : 0.f;
      bf[t] = (oc < OUT) ? W[(size_t)oc * IN + c] : 0.f;
    }
    acc = wmma4(af, bf, acc);
  }
#pragma unroll
  for (int r = 0; r < 8; ++r) {
    int m = bt * 16 + r + ((lane < 16) ? 0 : 8);
    float v = acc[r];
    if (bias) v += bias[oc];
    if (bns) v = v * bns[oc] + bnh[oc];
    if (do_lrelu) v = lrelu(v);
    if (m < Brows && oc < OUT) Y[(size_t)m * OUT + oc] = v;
  }
}

// ------------------------------ host sequence ------------------------------

extern "C" void kernel_launch(void* const* d_in, const int* in_sizes, int n_in,
                              void* d_out, int out_size, void* d_ws, size_t ws_size,
                              hipStream_t stream) {
  (void)in_sizes; (void)n_in; (void)out_size; (void)ws_size;

  const float* input = (const float*)d_in[0];
  // d_in[1] cloud_len_list, d_in[2] voxel_num: unused by the reference math.
  const float* W1 = (const float*)d_in[3];
  const float* W2 = (const float*)d_in[4];
  const float* W3 = (const float*)d_in[5];
  const float* W4 = (const float*)d_in[6];
  const float* W5 = (const float*)d_in[7];
  const float* g5 = (const float*)d_in[8];  const float* b5 = (const float*)d_in[9];
  const float* m5 = (const float*)d_in[10]; const float* v5 = (const float*)d_in[11];
  const float* lin1W = (const float*)d_in[12];
  const float* g6 = (const float*)d_in[13]; const float* b6 = (const float*)d_in[14];
  const float* m6 = (const float*)d_in[15]; const float* v6 = (const float*)d_in[16];
  const float* lin2W = (const float*)d_in[17];
  const float* lin2b = (const float*)d_in[18];
  const float* g7 = (const float*)d_in[19]; const float* b7 = (const float*)d_in[20];
  const float* m7 = (const float*)d_in[21]; const float* v7 = (const float*)d_in[22];
  const float* lin3W = (const float*)d_in[23];
  const float* lin3b = (const float*)d_in[24];
  const float* W6 = (const float*)d_in[25];
  const float* g6c = (const float*)d_in[26]; const float* b6c = (const float*)d_in[27];
  const float* m6c = (const float*)d_in[28]; const float* v6c = (const float*)d_in[29];
  const float* W7 = (const float*)d_in[30];
  const float* g7c = (const float*)d_in[31]; const float* b7c = (const float*)d_in[32];
  const float* m7c = (const float*)d_in[33]; const float* v7c = (const float*)d_in[34];
  const float* W8 = (const float*)d_in[35];
  const float* g8c = (const float*)d_in[36]; const float* b8c = (const float*)d_in[37];
  const float* m8c = (const float*)d_in[38]; const float* v8c = (const float*)d_in[39];
  const float* lin4W = (const float*)d_in[40];
  const float* lin5W = (const float*)d_in[41];
  const float* lin5b = (const float*)d_in[42];
  float* out = (float*)d_out;

  const int B = 64, N = 512, K = 16;
  const int B2 = 2, V = 32, VK = 8, CLS = 40, EMB = 1024;

  // ---- workspace carve (floats; int regions reuse 4-byte slots) ----
  float* ws = (float*)d_ws;
  size_t off = 0;
  float* x0 = ws + off;    off += (size_t)B * 9 * N;      // (64,9,512)
  float* xc = ws + off;    off += (size_t)B * 256 * N;    // (64,256,512) concat buf
  float* xx = ws + off;    off += (size_t)B * N;          // sq-norms
  int* idxA = (int*)(ws + off); off += (size_t)B * N * K; // (64,512,16)
  float* feat = ws + off;  off += (size_t)B * 2 * EMB;    // (64,2048)
  float* h1 = ws + off;    off += (size_t)B * 1024;
  float* h2 = ws + off;    off += (size_t)B * 1024;
  float* cls = ws + off;   off += (size_t)B * CLS;
  float* y0 = ws + off;    off += (size_t)B2 * CLS * V;   // (2,40,32)
  float* yxx = ws + off;   off += (size_t)B2 * V;
  int* idxB = (int*)(ws + off); off += (size_t)B2 * V * VK;
  float* yc = ws + off;    off += (size_t)B2 * 768 * V;   // (2,768,32) concat buf
  float* feat2 = ws + off; off += (size_t)B2 * 2 * EMB;
  float* h4 = ws + off;    off += (size_t)B2 * 256;
  float* bn5s = ws + off;  off += 1024;  float* bn5h = ws + off;  off += 1024;
  float* bn6s = ws + off;  off += 1024;  float* bn6h = ws + off;  off += 1024;
  float* bn7s = ws + off;  off += 1024;  float* bn7h = ws + off;  off += 1024;
  float* bn6cs = ws + off; off += 512;   float* bn6ch = ws + off; off += 512;
  float* bn7cs = ws + off; off += 256;   float* bn7ch = ws + off; off += 256;
  float* bn8cs = ws + off; off += 1024;  float* bn8ch = ws + off; off += 1024;

  // ---- fold batchnorms ----
  bnprep_kernel<<<4, 256, 0, stream>>>(g5, b5, m5, v5, 1024, bn5s, bn5h);
  bnprep_kernel<<<4, 256, 0, stream>>>(g6, b6, m6, v6, 1024, bn6s, bn6h);
  bnprep_kernel<<<4, 256, 0, stream>>>(g7, b7, m7, v7, 1024, bn7s, bn7h);
  bnprep_kernel<<<2, 256, 0, stream>>>(g6c, b6c, m6c, v6c, 512, bn6cs, bn6ch);
  bnprep_kernel<<<1, 256, 0, stream>>>(g7c, b7c, m7c, v7c, 256, bn7cs, bn7ch);
  bnprep_kernel<<<4, 256, 0, stream>>>(g8c, b8c, m8c, v8c, 1024, bn8cs, bn8ch);

  // ---- stage A: point cloud (B=64, N=512) ----
  {
    int tot = B * 9 * N;
    reshape_in_kernel<<<(tot + 255) / 256, 256, 0, stream>>>(input, x0, tot);
  }
  const int totBN = B * N;
  const int xcStride = 256 * N;

  // edge conv 1: x0(9ch) -> xc[0:32)
  sqnorm_kernel<<<(totBN + 255) / 256, 256, 0, stream>>>(x0, 9 * N, 9, N, xx, totBN);
  knn_kernel<16><<<(totBN + 255) / 256, 256, 0, stream>>>(x0, 9 * N, 9, N, xx, idxA, totBN);
  edgeconv_kernel<16><<<dim3(N, B), 32, 0, stream>>>(x0, 9 * N, 9, N, idxA, W1, 32,
                                                     nullptr, nullptr, xc, xcStride);
  // edge conv 2: x1(32ch) -> xc[32:64)
  sqnorm_kernel<<<(totBN + 255) / 256, 256, 0, stream>>>(xc, xcStride, 32, N, xx, totBN);
  knn_kernel<16><<<(totBN + 255) / 256, 256, 0, stream>>>(xc, xcStride, 32, N, xx, idxA, totBN);
  edgeconv_kernel<16><<<dim3(N, B), 32, 0, stream>>>(xc, xcStride, 32, N, idxA, W2, 32,
                                                     nullptr, nullptr, xc + 32 * N, xcStride);
  // edge conv 3: x2(32ch) -> xc[64:128)
  sqnorm_kernel<<<(totBN + 255) / 256, 256, 0, stream>>>(xc + 32 * N, xcStride, 32, N, xx, totBN);
  knn_kernel<16><<<(totBN + 255) / 256, 256, 0, stream>>>(xc + 32 * N, xcStride, 32, N, xx, idxA, totBN);
  edgeconv_kernel<16><<<dim3(N, B), 32, 0, stream>>>(xc + 32 * N, xcStride, 32, N, idxA,
                                                     W3, 64, nullptr, nullptr,
                                                     xc + 64 * N, xcStride);
  // edge conv 4: x3(64ch) -> xc[128:256)
  sqnorm_kernel<<<(totBN + 255) / 256, 256, 0, stream>>>(xc + 64 * N, xcStride, 64, N, xx, totBN);
  knn_kernel<16><<<(totBN + 255) / 256, 256, 0, stream>>>(xc + 64 * N, xcStride, 64, N, xx, idxA, totBN);
  edgeconv_kernel<16><<<dim3(N, B), 32, 0, stream>>>(xc + 64 * N, xcStride, 64, N, idxA,
                                                     W4, 128, nullptr, nullptr,
                                                     xc + 128 * N, xcStride);

  // conv5 (256->1024) + bn + lrelu + fused max/mean pool -> feat (64,2048)
  convpool_kernel<<<dim3(EMB / 16, B), 32, 0, stream>>>(xc, 256, N, W5, bn5s, bn5h,
                                                        feat, EMB);

  // FC head
  linear_kernel<<<dim3(1024 / 16, B / 16), 32, 0, stream>>>(feat, B, 2 * EMB, lin1W,
                                                            nullptr, bn6s, bn6h, 1, h1, 1024);
  linear_kernel<<<dim3(1024 / 16, B / 16), 32, 0, stream>>>(h1, B, 1024, lin2W, lin2b,
                                                            bn7s, bn7h, 1, h2, 1024);
  linear_kernel<<<dim3((CLS + 15) / 16, B / 16), 32, 0, stream>>>(h2, B, 1024, lin3W,
                                                                  lin3b, nullptr, nullptr,
                                                                  0, cls, CLS);

  // ---- stage B: voxel graph (B2=2, V=32, VK=8) ----
  {
    int tot = B2 * CLS * V;
    reshape_cls_kernel<<<(tot + 255) / 256, 256, 0, stream>>>(cls, y0, tot);
  }
  const int totB2V = B2 * V;
  const int ycStride = 768 * V;

  // edge conv 6: y0(40ch) -> yc[0:512)
  sqnorm_kernel<<<1, 256, 0, stream>>>(y0, CLS * V, CLS, V, yxx, totB2V);
  knn_kernel<8><<<1, 256, 0, stream>>>(y0, CLS * V, CLS, V, yxx, idxB, totB2V);
  edgeconv_kernel<8><<<dim3(V, B2), 32, 0, stream>>>(y0, CLS * V, CLS, V, idxB, W6, 512,
                                                     bn6cs, bn6ch, yc, ycStride);
  // edge conv 7: y1(512ch) -> yc[512:768)
  sqnorm_kernel<<<1, 256, 0, stream>>>(yc, ycStride, 512, V, yxx, totB2V);
  knn_kernel<8><<<1, 256, 0, stream>>>(yc, ycStride, 512, V, yxx, idxB, totB2V);
  edgeconv_kernel<8><<<dim3(V, B2), 32, 0, stream>>>(yc, ycStride, 512, V, idxB, W7, 256,
                                                     bn7cs, bn7ch, yc + 512 * V, ycStride);

  // conv8 (768->1024) + bn + lrelu + fused pool -> feat2 (2,2048)
  convpool_kernel<<<dim3(EMB / 16, B2), 32, 0, stream>>>(yc, 768, V, W8, bn8cs, bn8ch,
                                                         feat2, EMB);

  // final FCs -> out (2,40)
  linear_kernel<<<dim3(256 / 16, 1), 32, 0, stream>>>(feat2, B2, 2 * EMB, lin4W, nullptr,
                                                      nullptr, nullptr, 1, h4, 256);
  linear_kernel<<<dim3((40 + 15) / 16, 1), 32, 0, stream>>>(h4, B2, 256, lin5W, lin5b,
                                                            nullptr, nullptr, 0, out, 40);
}